// MarginalValueTheoremModel_31284541784591
// MI455X (gfx1250) — compile-verified
//
#include <hip/hip_runtime.h>
#include <hip/hip_bf16.h>
#include <math.h>

// Problem dims (fixed by the reference)
#define T_DIM 2048
#define B_DIM 4096
#define P_DIM 512
#define REC   7                      // floats per (t,b) record
#define CHUNK 64                     // timesteps per chunk
#define NC    (T_DIM / CHUNK)        // 32 chunks
#define BLK   256                    // threads per block (8 waves of 32)
#define NBT   (B_DIM / BLK)          // 16 column tiles
#define STAGE 4                      // timesteps staged per async batch
#define NSTG  (CHUNK / STAGE)        // 16 stages per chunk
#define PER_STEP_WORDS (BLK * REC)   // 1792 words per staged timestep
#define STG_WORDS (STAGE * PER_STEP_WORDS) // 7168 words per stage
#define STEP_CHUNKS (PER_STEP_WORDS / 4)   // 448 16-byte chunks per step
#define STG_CHUNKS  (STG_WORDS / 4)        // 1792 16-byte chunks per stage

#if defined(__HIP_DEVICE_COMPILE__) && \
    __has_builtin(__builtin_amdgcn_global_load_async_to_lds_b128) && \
    __has_builtin(__builtin_amdgcn_s_wait_asynccnt)
#define ASYNC_MODE 2                 // b128 copies (512 B / wave-instruction)
#elif defined(__HIP_DEVICE_COMPILE__) && \
    __has_builtin(__builtin_amdgcn_global_load_async_to_lds_b32) && \
    __has_builtin(__builtin_amdgcn_s_wait_asynccnt)
#define ASYNC_MODE 1                 // b32 copies (verified signature)
#else
#define ASYNC_MODE 0                 // synchronous fallback
#endif

#if ASYNC_MODE
// b32 signature confirmed by clang diagnostic: (int* AS1, int* AS3, Imm, Imm)
typedef __attribute__((address_space(1))) int gas_int;
typedef __attribute__((address_space(3))) int lds_int;
typedef int v4i_ __attribute__((ext_vector_type(4)));
typedef __attribute__((address_space(1))) v4i_ gas_v4i;
typedef __attribute__((address_space(3))) v4i_ lds_v4i;
#if ASYNC_MODE == 2
#define ISSUES_PER_STAGE (STG_CHUNKS / BLK)      // 7 b128 copies per thread
#else
#define ISSUES_PER_STAGE (STAGE * REC)           // 28 b32 copies per thread
#endif
#endif

struct Params { float la2, beta, c, depl, base; };

__device__ __forceinline__ float softplusf_(float x) {
  return (x > 20.0f) ? x : log1pf(expf(x));
}
__device__ __forceinline__ float clipf_(float x, float lo, float hi) {
  return fminf(fmaxf(x, lo), hi);
}

__device__ __forceinline__ Params load_params(
    const float* __restrict__ x, int b,
    const float* __restrict__ aR, const float* __restrict__ bR,
    const float* __restrict__ cR, const float* __restrict__ dR,
    const float* __restrict__ gR) {
  int pid = (int)x[(size_t)b * REC + 6];        // x[0, b, 6]
  float alpha = clipf_(1.0f / (1.0f + expf(-aR[pid])), 0.01f, 0.99f);
  Params p;
  p.la2  = log2f(1.0f - alpha);                 // (1-a)^ts = exp2(ts*la2)
  p.beta = clipf_(softplusf_(bR[pid]), 0.1f, 10.0f);
  p.c    = clipf_(cR[pid], -10.0f, 10.0f);
  p.depl = clipf_(softplusf_(dR[pid]), 0.001f, 1.0f);
  p.base = clipf_(softplusf_(gR[pid]), 0.1f, 20.0f);
  return p;
}

// ---------------------------------------------------------------------------
// Pass 1: per (chunk, column) compose chunk-level affine summaries:
//   egr' = Ac*egr + Bc ;  (cum/nh/tip)' = m*(state) + s  (reset-affine)
// ---------------------------------------------------------------------------
__global__ void __launch_bounds__(BLK) mvt_pass1(
    const float* __restrict__ x,
    const float* __restrict__ aR, const float* __restrict__ bR,
    const float* __restrict__ cR, const float* __restrict__ dR,
    const float* __restrict__ gR,
    float* __restrict__ ws) {
  const int chunk = blockIdx.x / NBT;
  const int b = (blockIdx.x % NBT) * BLK + threadIdx.x;
  const Params p = load_params(x, b, aR, bR, cR, dR, gR);

  float Ac = 1.0f, Bc = 0.0f;
  float m = 1.0f, sc = 0.0f, sn = 0.0f, st = 0.0f;
  const int t0 = chunk * CHUNK;
  const size_t stride = (size_t)B_DIM * REC;
  const float* rec = x + (size_t)t0 * stride + (size_t)b * REC;

  for (int t = 0; t < CHUNK; ++t) {
    if (t0 + t + 8 < T_DIM) __builtin_prefetch(rec + 8 * stride, 0, 0);
    float did = rec[0];
    float ar  = rec[2] + rec[3];
    float hd  = rec[4];
    float td  = rec[5];
    float tsv = hd * did + td * (1.0f - did);
    float At  = exp2f(tsv * p.la2);
    float obs = ar / (tsv + 1e-8f);
    Bc = At * Bc + (1.0f - At) * obs;
    Ac *= At;
    if (did > 0.5f) { sc += ar; sn += 1.0f; st += tsv; }
    else            { m = 0.0f; sc = 0.0f; sn = 0.0f; st = 0.0f; }
    rec += stride;
  }
  const int N = NC * B_DIM;
  const int idx = chunk * B_DIM + b;
  ws[0 * N + idx] = Ac;
  ws[1 * N + idx] = Bc;
  ws[2 * N + idx] = m;
  ws[3 * N + idx] = sc;
  ws[4 * N + idx] = sn;
  ws[5 * N + idx] = st;
}

// ---------------------------------------------------------------------------
// Pass 2: per column, chain the 32 chunk summaries; record egr at each chunk
// entry; emit the 5 final scalars (cum, nh, tip, egr, cts).
// ---------------------------------------------------------------------------
__global__ void __launch_bounds__(BLK) mvt_pass2(
    const float* __restrict__ x,
    const float* __restrict__ aR, const float* __restrict__ bR,
    const float* __restrict__ cR, const float* __restrict__ dR,
    const float* __restrict__ gR,
    float* __restrict__ ws, float* __restrict__ out) {
  const int b = blockIdx.x * BLK + threadIdx.x;
  const Params p = load_params(x, b, aR, bR, cR, dR, gR);
  const int N = NC * B_DIM;

  float egr = p.base, cum = 0.0f, nh = 0.0f, tip = 0.0f;
  for (int c = 0; c < NC; ++c) {
    const int idx = c * B_DIM + b;
    ws[6 * N + idx] = egr;                      // egr at chunk entry
    float Ac = ws[0 * N + idx], Bc = ws[1 * N + idx];
    float mm = ws[2 * N + idx];
    float sc = ws[3 * N + idx], sn = ws[4 * N + idx], st = ws[5 * N + idx];
    egr = Ac * egr + Bc;
    cum = mm * cum + sc;
    nh  = mm * nh  + sn;
    tip = mm * tip + st;
  }
  // final cts depends only on the last record
  const float* last = x + ((size_t)(T_DIM - 1) * B_DIM + b) * REC;
  float did = last[0];
  float ar  = last[2] + last[3];
  float cts = (did > 0.5f) ? ar : p.base;

  const size_t L = (size_t)T_DIM * B_DIM * 2;
  out[L + 0 * B_DIM + b] = cum;
  out[L + 1 * B_DIM + b] = nh;
  out[L + 2 * B_DIM + b] = tip;
  out[L + 3 * B_DIM + b] = egr;
  out[L + 4 * B_DIM + b] = cts;
}

// ---------------------------------------------------------------------------
// Pass 3: replay each chunk with the true entry egr; stream logits out.
// Input records are staged into LDS via async global->LDS copies (double
// buffered, tracked with ASYNCcnt) so the recurrence math overlaps the DMA.
// ---------------------------------------------------------------------------
__global__ void __launch_bounds__(BLK) mvt_pass3(
    const float* __restrict__ x,
    const float* __restrict__ aR, const float* __restrict__ bR,
    const float* __restrict__ cR, const float* __restrict__ dR,
    const float* __restrict__ gR,
    const float* __restrict__ ws, float* __restrict__ out) {
  const int chunk = blockIdx.x / NBT;
  const int bBase = (blockIdx.x % NBT) * BLK;
  const int tid = threadIdx.x;
  const int b = bBase + tid;
  const Params p = load_params(x, b, aR, bR, cR, dR, gR);
  const int N = NC * B_DIM;
  const int t0 = chunk * CHUNK;

  float egr = ws[6 * N + chunk * B_DIM + b];
  float cts;
  if (chunk == 0) {
    cts = p.base;
  } else {
    const float* pr = x + ((size_t)(t0 - 1) * B_DIM + b) * REC;
    float d = pr[0];
    float ar = pr[2] + pr[3];
    cts = (d > 0.5f) ? ar : p.base;
  }

#if ASYNC_MODE
  __shared__ float lds[2 * STG_WORDS];          // 57,344 bytes
  auto issue_stage = [&](int k) {
    const int kt = t0 + k * STAGE;
    float* dstbase = &lds[(k & 1) * STG_WORDS];
#if ASYNC_MODE == 2
    // 1792 16-byte chunks per stage; 7 per thread. Step bases are 16B-aligned
    // (multiples of 7168 B in both global and LDS).
#pragma unroll
    for (int i = 0; i < ISSUES_PER_STAGE; ++i) {
      const int j = i * BLK + tid;              // chunk id within stage
      const int s = j / STEP_CHUNKS;            // which staged timestep
      const int c = j - s * STEP_CHUNKS;        // 16B chunk within the step
      const float* g = x + ((size_t)(kt + s) * B_DIM + bBase) * REC + c * 4;
      float* d = dstbase + s * PER_STEP_WORDS + c * 4;
      __builtin_amdgcn_global_load_async_to_lds_b128(
          (gas_v4i*)g, (lds_v4i*)d, 0, 0);
    }
#else
    for (int s = 0; s < STAGE; ++s) {
      const float* g = x + ((size_t)(kt + s) * B_DIM + bBase) * REC;
      float* d = dstbase + s * PER_STEP_WORDS;
#pragma unroll
      for (int i = 0; i < REC; ++i) {
        const int w = i * BLK + tid;
        __builtin_amdgcn_global_load_async_to_lds_b32(
            (gas_int*)(g + w), (lds_int*)(d + w), 0, 0);
      }
    }
#endif
  };
  issue_stage(0);
  for (int k = 0; k < NSTG; ++k) {
    if (k + 1 < NSTG) {
      issue_stage(k + 1);
      // Newer stage's copies may remain outstanding; stage k's are complete.
      __builtin_amdgcn_s_wait_asynccnt(ISSUES_PER_STAGE);
    } else {
      __builtin_amdgcn_s_wait_asynccnt(0);
    }
    __syncthreads();
    const float* sbuf = &lds[(k & 1) * STG_WORDS];
    for (int s = 0; s < STAGE; ++s) {
      const float* rec = sbuf + s * PER_STEP_WORDS + tid * REC;
      float did = rec[0];
      float r0 = rec[2], r1 = rec[3];
      float hd = rec[4], td = rec[5];
      float logit0 = p.c + p.beta * (cts * p.depl - egr * hd);
      const int t = t0 + k * STAGE + s;
      *(float2*)(out + ((size_t)t * B_DIM + b) * 2) = make_float2(logit0, 0.0f);
      float ar  = r0 + r1;
      float tsv = hd * did + td * (1.0f - did);
      float At  = exp2f(tsv * p.la2);
      egr = At * egr + (1.0f - At) * (ar / (tsv + 1e-8f));
      cts = (did > 0.5f) ? ar : p.base;
    }
    __syncthreads();
  }
#else
  const size_t stride = (size_t)B_DIM * REC;
  const float* rec = x + (size_t)t0 * stride + (size_t)b * REC;
  for (int t = 0; t < CHUNK; ++t) {
    float did = rec[0];
    float r0 = rec[2], r1 = rec[3];
    float hd = rec[4], td = rec[5];
    float logit0 = p.c + p.beta * (cts * p.depl - egr * hd);
    *(float2*)(out + ((size_t)(t0 + t) * B_DIM + b) * 2) =
        make_float2(logit0, 0.0f);
    float ar  = r0 + r1;
    float tsv = hd * did + td * (1.0f - did);
    float At  = exp2f(tsv * p.la2);
    egr = At * egr + (1.0f - At) * (ar / (tsv + 1e-8f));
    cts = (did > 0.5f) ? ar : p.base;
    rec += stride;
  }
#endif
}

extern "C" void kernel_launch(void* const* d_in, const int* in_sizes, int n_in,
                              void* d_out, int out_size, void* d_ws, size_t ws_size,
                              hipStream_t stream) {
  const float* x  = (const float*)d_in[0];
  const float* aR = (const float*)d_in[1];  // alpha_env_raw  [P]
  const float* bR = (const float*)d_in[2];  // beta_raw       [P]
  const float* cR = (const float*)d_in[3];  // c_raw          [P]
  const float* dR = (const float*)d_in[4];  // depletion_raw  [P]
  const float* gR = (const float*)d_in[5];  // baseline_gain  [P]
  float* out = (float*)d_out;
  float* ws  = (float*)d_ws;                // needs 7*NC*B floats (~3.7 MB)

  const dim3 blk(BLK);
  const dim3 gridCB(NC * NBT);              // 512 blocks: (chunk, col-tile)
  mvt_pass1<<<gridCB, blk, 0, stream>>>(x, aR, bR, cR, dR, gR, ws);
  mvt_pass2<<<dim3(B_DIM / BLK), blk, 0, stream>>>(x, aR, bR, cR, dR, gR, ws, out);
  mvt_pass3<<<gridCB, blk, 0, stream>>>(x, aR, bR, cR, dR, gR, ws, out);
}